// DiffusionBonds_82841329205437
// MI455X (gfx1250) — compile-verified
//
#include <hip/hip_runtime.h>

typedef float v2f __attribute__((ext_vector_type(2)));
typedef float v8f __attribute__((ext_vector_type(8)));

#define LEAKY 0.001f
#define T_STEPS 8

__device__ __forceinline__ v8f wmma_f32(v2f a, v2f b, v8f c) {
    // D = A(16x4,f32) * B(4x16,f32) + C(16x16,f32)
    return __builtin_amdgcn_wmma_f32_16x16x4_f32(false, a, false, b, (short)0, c, false, false);
}
__device__ __forceinline__ float lrelu(float x) { return fmaxf(x, LEAKY * x); }

// ---- LDS layout (units: floats/dwords) ----
// W2/W3 stored pair-interleaved: pair-row r2 holds rows (2*r2, 2*r2+1) as float2
// per column.  Pair-row stride 288 dwords => half-wave offset == 32 mod 64 banks
// => conflict-free ds_load_b64 B-fragments.
constexpr int SWP  = 288;  // dwords per pair-row of W2/W3 (256 data + 32 pad)
constexpr int SW4P = 32;   // dwords per pair-row of W4 (16 cols, no pad needed)
constexpr int SH   = 132;  // padded row stride for 16x128 activation tiles

constexpr int LW2_OFF = 0;
constexpr int LW3_OFF = LW2_OFF + 64 * SWP;      // 18432
constexpr int LW4_OFF = LW3_OFF + 64 * SWP;      // 36864
constexpr int LB1_OFF = LW4_OFF + 64 * SW4P;     // +2048
constexpr int LB2_OFF = LB1_OFF + 128;
constexpr int LB3_OFF = LB2_OFF + 128;
constexpr int LT_OFF  = LB3_OFF + 128;   // W1 row 256 (t weight)
constexpr int LDL_OFF = LT_OFF + 128;    // W1 row 257 (dl weight)
constexpr int LB4_OFF = LDL_OFF + 128;   // 2 floats (+pad)
constexpr int WAVE_OFF = LB4_OFF + 16;

constexpr int PW_H     = 0;              // 16 x SH activation buffer A
constexpr int PW_H2    = PW_H + 16 * SH; // 16 x SH activation buffer B
constexpr int PW_DL    = PW_H2 + 16 * SH;   // 16 bond lengths
constexpr int PW_DH    = PW_DL + 16;        // 16 x 3 unit vectors
constexpr int PW_DELTA = PW_DH + 48;        // 16 x 2 deltas
constexpr int PW_I0    = PW_DELTA + 32;     // 16 ints
constexpr int PW_I1    = PW_I0 + 16;        // 16 ints
constexpr int PW_SIZE  = PW_I1 + 16;        // 4352 floats per wave

constexpr int WAVES = 8;
constexpr int SMEM_FLOATS = WAVE_OFF + WAVES * PW_SIZE;   // ~74384 floats = ~291 KB

// dense 128->128 layer with leaky relu: Hin (16xSH LDS) -> Hout (16xSH LDS)
// smemW points at a pair-interleaved 64xSWP weight tile.
__device__ __forceinline__ void layer128(const float* __restrict__ Hin,
                                         float* __restrict__ Hout,
                                         const float* __restrict__ smemW,
                                         const float* __restrict__ smemB,
                                         int h, int ln) {
    v8f c[8] = {};
    const float* WB = smemW + h * SWP + 2 * ln;      // pair-row 2k+h, col 16j+ln
    const float* HA = Hin + ln * SH + 2 * h;         // A[ln][4k+2h]
#pragma unroll 4
    for (int k = 0; k < 32; ++k) {
        v2f a = *reinterpret_cast<const v2f*>(HA + 4 * k);
#pragma unroll
        for (int j = 0; j < 8; ++j) {
            v2f b = *reinterpret_cast<const v2f*>(WB + (2 * k) * SWP + 32 * j);
            c[j] = wmma_f32(a, b, c[j]);
        }
    }
#pragma unroll
    for (int j = 0; j < 8; ++j) {
        const float bn = smemB[16 * j + ln];
#pragma unroll
        for (int v = 0; v < 8; ++v) {
            float x = c[j][v] + bn;
            Hout[(v + 8 * h) * SH + 16 * j + ln] = lrelu(x);
        }
    }
}

__global__ void __launch_bounds__(256, 1)
DiffusionBonds_kernel(const float* __restrict__ coords,
                      const float* __restrict__ encoded,
                      const float* __restrict__ tarr,
                      const float* __restrict__ W1, const float* __restrict__ b1,
                      const float* __restrict__ W2, const float* __restrict__ b2,
                      const float* __restrict__ W3, const float* __restrict__ b3,
                      const float* __restrict__ W4, const float* __restrict__ b4,
                      const int* __restrict__ bonds,
                      float* __restrict__ out,
                      int E) {
    extern __shared__ float smem[];
    const int tid = threadIdx.x;

    // ---- cooperative weight staging into LDS (pair-interleaved) ----
    for (int i = tid; i < 128 * 128; i += 256) {
        int r = i >> 7, n = i & 127;
        int dst = (r >> 1) * SWP + 2 * n + (r & 1);
        smem[LW2_OFF + dst] = W2[i];
        smem[LW3_OFF + dst] = W3[i];
    }
    for (int i = tid; i < 64 * SW4P; i += 256) {
        int r2 = i >> 5, q = i & 31;
        int n = q >> 1, half = q & 1;
        int row = 2 * r2 + half;
        smem[LW4_OFF + i] = (n < 2) ? W4[row * 2 + n] : 0.0f;
    }
    if (tid < 128) {
        smem[LB1_OFF + tid] = b1[tid];
        smem[LB2_OFF + tid] = b2[tid];
        smem[LB3_OFF + tid] = b3[tid];
        smem[LT_OFF + tid]  = W1[256 * 128 + tid];
        smem[LDL_OFF + tid] = W1[257 * 128 + tid];
    }
    if (tid < 2) smem[LB4_OFF + tid] = b4[tid];
    __syncthreads();

    const int lane = tid & 31;
    const int wv = tid >> 5;
    const int h = lane >> 4;    // half-wave selector (A/B frag K split)
    const int ln = lane & 15;   // row / column within tile

    float* WS = smem + WAVE_OFF + wv * PW_SIZE;
    float* H = WS + PW_H;
    float* H2 = WS + PW_H2;
    float* Sdl = WS + PW_DL;
    float* Sdh = WS + PW_DH;
    float* Sdelta = WS + PW_DELTA;
    int* Si0 = (int*)(WS + PW_I0);
    int* Si1 = (int*)(WS + PW_I1);

    const int tileBase = (blockIdx.x * WAVES + wv) * 16;   // 16 bonds per wave
    if (tileBase >= E) return;

    // ---- per-bond geometry (both half-waves compute the same 16 bonds) ----
    const int bidx = min(tileBase + ln, E - 1);
    const int i0 = bonds[2 * bidx];
    const int i1 = bonds[2 * bidx + 1];
    float dx = coords[3 * i0 + 0] - coords[3 * i1 + 0];
    float dy = coords[3 * i0 + 1] - coords[3 * i1 + 1];
    float dz = coords[3 * i0 + 2] - coords[3 * i1 + 2];
    float d2 = fmaxf(dx * dx + dy * dy + dz * dz, 1e-12f);
    float dl = sqrtf(d2);
    float inv = 1.0f / dl;
    if (lane < 16) {
        Sdl[ln] = dl;
        Sdh[ln * 3 + 0] = dx * inv;
        Sdh[ln * 3 + 1] = dy * inv;
        Sdh[ln * 3 + 2] = dz * inv;
        Si0[ln] = i0;
        Si1[ln] = i1;
    }

    // ---- layer 1 (t-independent part), accumulators held in VGPRs ----
    v8f g[8] = {};
    const float* A0 = encoded + i0 * 128 + 2 * h;
    const float* A1 = encoded + i1 * 128 + 2 * h;
    const float* Bp = W1 + (2 * h) * 128 + ln;   // W1[4k+2h][16j+ln]
#pragma unroll 4
    for (int k = 0; k < 32; ++k) {               // K = 0..127 : encoded[i0]
        v2f a = *reinterpret_cast<const v2f*>(A0 + 4 * k);
#pragma unroll
        for (int j = 0; j < 8; ++j) {
            v2f b;
            b.x = Bp[(4 * k) * 128 + 16 * j];
            b.y = Bp[(4 * k + 1) * 128 + 16 * j];
            g[j] = wmma_f32(a, b, g[j]);
        }
    }
#pragma unroll 4
    for (int k = 0; k < 32; ++k) {               // K = 128..255 : encoded[i1]
        v2f a = *reinterpret_cast<const v2f*>(A1 + 4 * k);
#pragma unroll
        for (int j = 0; j < 8; ++j) {
            v2f b;
            b.x = Bp[(128 + 4 * k) * 128 + 16 * j];
            b.y = Bp[(129 + 4 * k) * 128 + 16 * j];
            g[j] = wmma_f32(a, b, g[j]);
        }
    }

    // fold dl * W1[257] + b1 into g (pre-activation minus t term)
    float4 dla = *reinterpret_cast<const float4*>(Sdl + 8 * h);
    float4 dlb = *reinterpret_cast<const float4*>(Sdl + 8 * h + 4);
    float dlv[8] = {dla.x, dla.y, dla.z, dla.w, dlb.x, dlb.y, dlb.z, dlb.w};
#pragma unroll
    for (int j = 0; j < 8; ++j) {
        const float b1n = smem[LB1_OFF + 16 * j + ln];
        const float wdl = smem[LDL_OFF + 16 * j + ln];
#pragma unroll
        for (int v = 0; v < 8; ++v)
            g[j][v] = g[j][v] + b1n + dlv[v] * wdl;
    }

    // ---- loop over the 8 diffusion times ----
    for (int jt = 0; jt < T_STEPS; ++jt) {
        const float tv = tarr[jt];

        // h1 = lrelu(g + t * W1[256]) -> H (LDS, C-layout transpose)
#pragma unroll
        for (int j = 0; j < 8; ++j) {
            const float wt = smem[LT_OFF + 16 * j + ln];
#pragma unroll
            for (int v = 0; v < 8; ++v) {
                float x = fmaf(tv, wt, g[j][v]);
                H[(v + 8 * h) * SH + 16 * j + ln] = lrelu(x);
            }
        }

        layer128(H, H2, smem + LW2_OFF, smem + LB2_OFF, h, ln);   // layer 2
        layer128(H2, H, smem + LW3_OFF, smem + LB3_OFF, h, ln);   // layer 3

        // layer 4: [16,128] x [128,16(pad)] -> delta in cols 0,1
        v8f c4 = {};
        const float* WB4 = smem + LW4_OFF + h * SW4P + 2 * ln;
        const float* HA3 = H + ln * SH + 2 * h;
#pragma unroll 4
        for (int k = 0; k < 32; ++k) {
            v2f a = *reinterpret_cast<const v2f*>(HA3 + 4 * k);
            v2f b = *reinterpret_cast<const v2f*>(WB4 + (2 * k) * SW4P);
            c4 = wmma_f32(a, b, c4);
        }
        const float b4n = (ln < 2) ? smem[LB4_OFF + ln] : 0.0f;
        if (ln < 2) {
#pragma unroll
            for (int v = 0; v < 8; ++v)
                Sdelta[(v + 8 * h) * 2 + ln] = c4[v] + b4n;
        }

        // scatter: 16 bonds * 2 endpoints * 3 comps = 96 atomics, 3 rounds of 32 lanes
#pragma unroll
        for (int r = 0; r < 3; ++r) {
            int idx = r * 32 + lane;
            int bnd = idx / 6;
            int rem = idx - bnd * 6;
            int ep = rem / 3;
            int comp = rem - ep * 3;
            if (tileBase + bnd < E) {
                float dlt = Sdelta[bnd * 2 + ep];
                float dhc = Sdh[bnd * 3 + comp];
                float val = (ep ? 0.5f : -0.5f) * dlt * dhc;
                int node = ep ? Si1[bnd] : Si0[bnd];
                unsafeAtomicAdd(&out[(node * T_STEPS + jt) * 3 + comp], val);
            }
        }
    }
}

extern "C" void kernel_launch(void* const* d_in, const int* in_sizes, int n_in,
                              void* d_out, int out_size, void* d_ws, size_t ws_size,
                              hipStream_t stream) {
    const float* coords  = (const float*)d_in[0];
    const float* encoded = (const float*)d_in[1];
    const float* tarr    = (const float*)d_in[2];
    const float* answer  = (const float*)d_in[3];
    const float* W1      = (const float*)d_in[4];
    const float* b1      = (const float*)d_in[5];
    const float* W2      = (const float*)d_in[6];
    const float* b2      = (const float*)d_in[7];
    const float* W3      = (const float*)d_in[8];
    const float* b3      = (const float*)d_in[9];
    const float* W4      = (const float*)d_in[10];
    const float* b4      = (const float*)d_in[11];
    const int* bonds     = (const int*)d_in[12];
    float* out           = (float*)d_out;

    const int E = in_sizes[12] / 2;

    // out = answer (then kernel atomically accumulates updates)
    hipMemcpyAsync(d_out, (const void*)answer, (size_t)out_size * sizeof(float),
                   hipMemcpyDeviceToDevice, stream);

    const int tiles = (E + 15) / 16;
    const int blocks = (tiles + WAVES - 1) / WAVES;
    const size_t smemBytes = (size_t)SMEM_FLOATS * sizeof(float);

    hipFuncSetAttribute((const void*)DiffusionBonds_kernel,
                        hipFuncAttributeMaxDynamicSharedMemorySize, (int)smemBytes);

    DiffusionBonds_kernel<<<blocks, 256, smemBytes, stream>>>(
        coords, encoded, tarr, W1, b1, W2, b2, W3, b3, W4, b4, bonds, out, E);
}